// VectorQuantizer_68547678044279
// MI455X (gfx1250) — compile-verified
//
#include <hip/hip_runtime.h>

typedef float v2f __attribute__((ext_vector_type(2)));
typedef float v4f __attribute__((ext_vector_type(4)));
typedef float v8f __attribute__((ext_vector_type(8)));

#define HW    4096            // 64*64 spatial
#define DDIM  64              // embedding dim
#define KCB   1024            // codebook entries
#define BATCH 32
#define NVEC  (BATCH * HW)    // 131072 vectors
#define OUT_Q (BATCH * DDIM * HW)  // 8388608 floats, then NVEC index floats

__global__ __launch_bounds__(256)
void vq_wmma_kernel(const float* __restrict__ x,
                    const float* __restrict__ cb,
                    float* __restrict__ outq,
                    float* __restrict__ outi)
{
    __shared__ int sIdx[8 * 16];

    const int lane = threadIdx.x & 31;
    const int wv   = threadIdx.x >> 5;          // wave in block (0..7)
    const int wave = blockIdx.x * 8 + wv;       // global wave id
    const int v0   = wave * 16;                 // first vector handled
    const int b    = v0 >> 12;                  // / 4096  (never crosses batch: 4096%16==0)
    const int n0   = v0 & 4095;

    const float* __restrict__ xb = x + (size_t)b * (DDIM * HW) + n0;

    const int mres = lane & 15;                 // M (for A/C) or N column (for B/C)
    const int kp   = lane >> 4;                 // half-wave: K-pair select

    // ---- Load A tile 16x64 in WMMA f32 16x16x4 A-layout (reused for all 64 N-tiles)
    // lane L, step kk: holds A[M=L%16][4kk + 2*(L/16) + {0,1}]
    v2f a[16];
#pragma unroll
    for (int kk = 0; kk < 16; ++kk) {
        const int d0 = 4 * kk + 2 * kp;
        a[kk].x = xb[(size_t)(d0    ) * HW + mres];
        a[kk].y = xb[(size_t)(d0 + 1) * HW + mres];
    }

    float best[8];
    int   bidx[8];
#pragma unroll
    for (int v = 0; v < 8; ++v) { best[v] = 3.4e38f; bidx[v] = 0; }

    // ---- Sweep 64 tiles of 16 codebook entries
    for (int t = 0; t < 64; ++t) {
        const int code = t * 16 + mres;                       // this lane's code column
        const float* __restrict__ cr = cb + code * DDIM + 2 * kp;

        v8f c = {0.f, 0.f, 0.f, 0.f, 0.f, 0.f, 0.f, 0.f};
        float bn = 0.f;                                       // partial ||e||^2 (this half)
#pragma unroll
        for (int kk = 0; kk < 16; ++kk) {
            v2f bv = *(const v2f*)(cr + 4 * kk);              // 8B-aligned float2
            bn += bv.x * bv.x + bv.y * bv.y;
            // v_wmma_f32_16x16x4_f32: (neg_a, A, neg_b, B, c_mod, C, reuse_a, reuse_b)
            c = __builtin_amdgcn_wmma_f32_16x16x4_f32(
                    false, a[kk], false, bv, (short)0, c, false, false);
        }
        const float cbn = bn + __shfl_xor(bn, 16, 32);        // full ||e||^2 for this column

        // C layout: vgpr v, lane L -> M = v + 8*(L/16), N = L%16 (== this lane's code)
#pragma unroll
        for (int v = 0; v < 8; ++v) {
            const float s = cbn - 2.0f * c[v];
            if (s < best[v]) { best[v] = s; bidx[v] = code; } // strict '<' keeps lowest index
        }
    }

    // ---- Argmin reduce across the 16 columns (lanes within each half)
#pragma unroll
    for (int v = 0; v < 8; ++v) {
#pragma unroll
        for (int mask = 1; mask <= 8; mask <<= 1) {
            const float os = __shfl_xor(best[v], mask, 32);
            const int   oi = __shfl_xor(bidx[v], mask, 32);
            if (os < best[v] || (os == best[v] && oi < bidx[v])) {
                best[v] = os; bidx[v] = oi;
            }
        }
    }

    // Lanes with mres==0 (lanes 0 and 16) hold rows 8*kp .. 8*kp+7
    if (mres == 0) {
#pragma unroll
        for (int v = 0; v < 8; ++v) {
            const int row = 8 * kp + v;
            outi[v0 + row]     = (float)bidx[v];              // indices output (as float)
            sIdx[wv * 16 + row] = bidx[v];
        }
    }
    __syncthreads();

    // ---- Quantized output: out[b][d][n0+m] = cb[idx[m]][d], float4-coalesced in m
    const int mq = (lane & 3) * 4;      // 4 rows per lane
    const int dl = lane >> 2;           // 8 d-values per iter across wave
    const int i0 = sIdx[wv * 16 + mq + 0];
    const int i1 = sIdx[wv * 16 + mq + 1];
    const int i2 = sIdx[wv * 16 + mq + 2];
    const int i3 = sIdx[wv * 16 + mq + 3];
    float* __restrict__ ob = outq + (size_t)b * (DDIM * HW) + n0 + mq;
#pragma unroll
    for (int dblk = 0; dblk < 8; ++dblk) {
        const int d = dblk * 8 + dl;
        v4f q;
        q.x = cb[i0 * DDIM + d];
        q.y = cb[i1 * DDIM + d];
        q.z = cb[i2 * DDIM + d];
        q.w = cb[i3 * DDIM + d];
        *(v4f*)(ob + (size_t)d * HW) = q;                     // 16B-aligned
    }
}

extern "C" void kernel_launch(void* const* d_in, const int* in_sizes, int n_in,
                              void* d_out, int out_size, void* d_ws, size_t ws_size,
                              hipStream_t stream)
{
    (void)in_sizes; (void)n_in; (void)out_size; (void)d_ws; (void)ws_size;
    const float* x  = (const float*)d_in[0];
    const float* cb = (const float*)d_in[1];
    float* outq = (float*)d_out;            // 8,388,608 floats (quantized)
    float* outi = (float*)d_out + OUT_Q;    // 131,072 index values
    const int blocks = NVEC / (16 * 8);     // 1024 blocks of 8 waves
    vq_wmma_kernel<<<blocks, 256, 0, stream>>>(x, cb, outq, outi);
}